// SurrogatePrecision_3066606649434
// MI455X (gfx1250) — compile-verified
//
#include <hip/hip_runtime.h>
#include <hip/hip_bf16.h>
#include <math.h>

typedef __attribute__((ext_vector_type(2))) float v2f;
typedef __attribute__((ext_vector_type(8))) float v8f;

#define BS     512
#define DIM    384
#define KTOP   5.0f
#define INV_T2 100.0f
#define BIGD   1.0e6f
#define LOG2E  1.44269504088896340736f

// Fast sigmoid on the CDNA5 TRANS pipe: sigmoid(x) = rcp(1 + exp2(-x*log2e)).
// v_exp_f32 / v_rcp_f32 are TRANS-class and co-execute with normal VALU.
__device__ __forceinline__ float sigmoid_fast(float x) {
    return __builtin_amdgcn_rcpf(1.0f + __builtin_amdgcn_exp2f(-x * LOG2E));
}

// ---------------- kernel 1: per-row sum of squares ----------------
// One wave32 per row; 8 waves / 256-thread block; 64 blocks.
__global__ __launch_bounds__(256) void k_rowsq(const float* __restrict__ emb,
                                               float* __restrict__ sq) {
    const int lane = threadIdx.x & 31;
    const int wave = threadIdx.x >> 5;
    const int row  = blockIdx.x * 8 + wave;
    const float* r = emb + row * DIM;
    float s = 0.f;
    #pragma unroll
    for (int t = 0; t < DIM / 32; ++t) {
        float v = r[lane + 32 * t];
        s += v * v;
    }
    #pragma unroll
    for (int o = 16; o > 0; o >>= 1) s += __shfl_down(s, o, 32);
    if (lane == 0) sq[row] = s;
}

// ---------------- kernel 2: fp32 WMMA Gram -> Euclidean distance ----------------
// Each wave computes one 16x16 tile of G = emb @ emb^T via V_WMMA_F32_16X16X4_F32,
// then fuses dist = sqrt(max(sq_i + sq_j - 2G, 1e-12)), diag -> BIG.
// A layout (16x4 f32): lanes 0-15 hold M=lane, K={0,1}; lanes 16-31 hold M, K={2,3}.
// B layout (4x16 f32): lanes 0-15 hold N=lane rows K={0,1}; lanes 16-31 rows K={2,3}.
__global__ __launch_bounds__(128) void k_dist(const float* __restrict__ emb,
                                              const float* __restrict__ sq,
                                              float* __restrict__ dist) {
    const int lane = threadIdx.x & 31;
    const int wave = threadIdx.x >> 5;
    const int i0 = blockIdx.x * 16;
    const int j0 = (blockIdx.y * 4 + wave) * 16;
    const int mn = lane & 15;       // M for A / N for B
    const int hp = lane >> 4;       // wave half
    const int kb = hp * 2;          // K sub-offset for this half

    const float* arow = emb + (i0 + mn) * DIM + kb;
    const float* brow = emb + (j0 + mn) * DIM + kb;

    v8f c = {};
    #pragma unroll 4
    for (int k = 0; k < DIM; k += 4) {
        v2f a = *(const v2f*)(arow + k);
        v2f b = *(const v2f*)(brow + k);
        // 8 args: (neg_a, A, neg_b, B, c_mod, C, reuse_a, reuse_b)
        c = __builtin_amdgcn_wmma_f32_16x16x4_f32(false, a, false, b,
                                                  (short)0, c, false, false);
    }

    const int j = j0 + mn;
    const float sqj = sq[j];
    #pragma unroll
    for (int vg = 0; vg < 8; ++vg) {
        const int i = i0 + vg + 8 * hp;      // C/D layout: VGPR vg = rows vg, vg+8
        float d2 = sq[i] + sqj - 2.0f * c[vg];
        float d  = __builtin_amdgcn_sqrtf(fmaxf(d2, 1e-12f));  // v_sqrt_f32
        if (i == j) d = BIGD;
        dist[i * BS + j] = d;
    }
}

// ---------------- kernel 3: soft rank + soft precision per query ----------------
// One block per query. LDS row holds dist pre-scaled by log2(e)/T2 so each of
// the 512^3 terms is: ds_load(broadcast) -> sub -> v_exp_f32 -> add -> v_rcp_f32.
// sigmoid((dj - dm)/T2) = rcp(1 + exp2((dm - dj) * INV_T2 * log2e)).
// Diagonal BIG scales to ~1.4e8 -> exp2 saturates to inf/0 -> sigmoid 0/1 exact.
__global__ __launch_bounds__(256) void k_rank(const float* __restrict__ dist,
                                              const int* __restrict__ labels,
                                              float* __restrict__ prec) {
    __shared__ float rowz[BS];     // pre-scaled distances
    __shared__ float s_num[256];
    __shared__ float s_gt[256];

    const int q = blockIdx.x;
    const int t = threadIdx.x;
    const float C = INV_T2 * LOG2E;
    rowz[t]       = dist[q * BS + t] * C;
    rowz[t + 256] = dist[q * BS + t + 256] * C;
    __syncthreads();

    const int lq = labels[q];
    const float a0 = rowz[t];
    const float a1 = rowz[t + 256];
    float r0 = 0.f, r1 = 0.f;
    #pragma unroll 4
    for (int m = 0; m < BS; ++m) {
        const float zm = rowz[m];
        r0 += __builtin_amdgcn_rcpf(1.0f + __builtin_amdgcn_exp2f(zm - a0));
        r1 += __builtin_amdgcn_rcpf(1.0f + __builtin_amdgcn_exp2f(zm - a1));
    }

    float num = 0.f, gt = 0.f;
    if (labels[t] == lq && t != q)               { gt += 1.f; num += sigmoid_fast(KTOP - r0); }  // T1 == 1
    if (labels[t + 256] == lq && (t + 256) != q) { gt += 1.f; num += sigmoid_fast(KTOP - r1); }

    s_num[t] = num; s_gt[t] = gt;
    __syncthreads();
    for (int o = 128; o > 0; o >>= 1) {
        if (t < o) { s_num[t] += s_num[t + o]; s_gt[t] += s_gt[t + o]; }
        __syncthreads();
    }
    if (t == 0) prec[q] = s_num[0] / fminf(s_gt[0], KTOP);  // 0/0 -> NaN matches reference
}

// ---------------- kernel 4: loss = 1 - mean(precision) ----------------
__global__ __launch_bounds__(256) void k_final(const float* __restrict__ prec,
                                               float* __restrict__ out) {
    __shared__ float s[256];
    const int t = threadIdx.x;
    s[t] = prec[t] + prec[t + 256];
    __syncthreads();
    for (int o = 128; o > 0; o >>= 1) {
        if (t < o) s[t] += s[t + o];
        __syncthreads();
    }
    if (t == 0) out[0] = 1.0f - s[0] * (1.0f / (float)BS);
}

extern "C" void kernel_launch(void* const* d_in, const int* in_sizes, int n_in,
                              void* d_out, int out_size, void* d_ws, size_t ws_size,
                              hipStream_t stream) {
    const float* emb    = (const float*)d_in[0];
    const int*   labels = (const int*)d_in[1];

    float* ws   = (float*)d_ws;
    float* sq   = ws;            // 512 floats
    float* prec = ws + 512;      // 512 floats
    float* dist = ws + 1024;     // 512*512 floats (~1MB)

    k_rowsq<<<dim3(BS / 8),           dim3(256), 0, stream>>>(emb, sq);
    k_dist <<<dim3(BS / 16, BS / 64), dim3(128), 0, stream>>>(emb, sq, dist);
    k_rank <<<dim3(BS),               dim3(256), 0, stream>>>(dist, labels, prec);
    k_final<<<dim3(1),                dim3(256), 0, stream>>>(prec, (float*)d_out);
}